// BoundedMultiResGrid_15968688406830
// MI455X (gfx1250) — compile-verified
//
#include <hip/hip_runtime.h>

// BoundedMultiResGrid: 4-level dense-grid trilinear interpolation.
// Inputs (setup_inputs order): x [N,3] f32, emb0 [16,16,16,2], emb1 [32^3,2],
// emb2 [64^3,2], emb3 [128^3,2] (all f32).
// Output: features [N,8] f32 followed by mask [N] (written as 0.0/1.0 f32).

typedef float v4f __attribute__((ext_vector_type(4)));

__device__ __forceinline__ void grid_coord(float c, int R, int& i0, float& f) {
    // c is already clamped to [0,1] -> u in [0, R-1], truncation == floor.
    float u = c * (float)(R - 1);
    int i = (int)u;
    if (i > R - 2) i = R - 2;
    f = u - (float)i;
    i0 = i;
}

template <int R>
__device__ __forceinline__ void trilerp(const float2* __restrict__ e,
                                        float cx, float cy, float cz,
                                        float2& out) {
    int ix, iy, iz;
    float fx, fy, fz;
    grid_coord(cx, R, ix, fx);
    grid_coord(cy, R, iy, fy);
    grid_coord(cz, R, iz, fz);
    // layout [R,R,R,2] viewed as float2[R*R*R]; z is innermost.
    int b00 = (ix * R + iy) * R + iz;   // (x0,y0,z0)
    int b01 = b00 + R;                  // (x0,y1,z0)
    int b10 = b00 + R * R;              // (x1,y0,z0)
    int b11 = b10 + R;                  // (x1,y1,z0)
    float2 c000 = e[b00],     c001 = e[b00 + 1];
    float2 c010 = e[b01],     c011 = e[b01 + 1];
    float2 c100 = e[b10],     c101 = e[b10 + 1];
    float2 c110 = e[b11],     c111 = e[b11 + 1];
    float wz = 1.0f - fz, wy = 1.0f - fy, wx = 1.0f - fx;
    float2 c00 = make_float2(c000.x * wz + c001.x * fz, c000.y * wz + c001.y * fz);
    float2 c01 = make_float2(c010.x * wz + c011.x * fz, c010.y * wz + c011.y * fz);
    float2 c10 = make_float2(c100.x * wz + c101.x * fz, c100.y * wz + c101.y * fz);
    float2 c11 = make_float2(c110.x * wz + c111.x * fz, c110.y * wz + c111.y * fz);
    float2 c0  = make_float2(c00.x * wy + c01.x * fy, c00.y * wy + c01.y * fy);
    float2 c1  = make_float2(c10.x * wy + c11.x * fy, c10.y * wy + c11.y * fy);
    out = make_float2(c0.x * wx + c1.x * fx, c0.y * wx + c1.y * fx);
}

__global__ __launch_bounds__(256) void multires_grid_kernel(
    const float* __restrict__ x,
    const float* __restrict__ emb0,
    const float* __restrict__ emb1,
    const float* __restrict__ emb2,
    const float* __restrict__ emb3,
    float* __restrict__ out, int n) {
    // Level-0 table: 16^3 * float2 = 32 KB, staged in LDS (320 KB/WGP budget).
    __shared__ __align__(16) float2 s_emb0[16 * 16 * 16];

    // --- CDNA5 async global->LDS DMA stage of emb0 (8 x b128 per thread) ---
    {
        // Low 32 bits of a generic pointer into the LDS aperture are the LDS
        // byte offset (ISA 10.2 aperture decode), which is what the async
        // instruction's LDS-address VGPR expects.
        unsigned lds_base = (unsigned)(unsigned long long)(&s_emb0[0]);
        unsigned long long gbase = (unsigned long long)emb0;
        unsigned t16 = threadIdx.x * 16u;
#pragma unroll
        for (int k = 0; k < 8; ++k) {
            unsigned off = t16 + (unsigned)k * 4096u;      // 256 thr * 16 B
            unsigned lds_addr = lds_base + off;
            asm volatile("global_load_async_to_lds_b128 %0, %1, %2 offset:0"
                         :
                         : "v"(lds_addr), "v"(off), "s"(gbase)
                         : "memory");
        }
        asm volatile("s_wait_asynccnt 0x0" ::: "memory");
    }
    __syncthreads();

    const float2* e1 = (const float2*)emb1;
    const float2* e2 = (const float2*)emb2;
    const float2* e3 = (const float2*)emb3;

    long long stride = (long long)gridDim.x * blockDim.x;
    for (long long i = (long long)blockIdx.x * blockDim.x + threadIdx.x; i < n;
         i += stride) {
        float xv = x[3 * i + 0];
        float yv = x[3 * i + 1];
        float zv = x[3 * i + 2];

        bool inb = (xv >= 0.0f) & (xv <= 1.0f) & (yv >= 0.0f) & (yv <= 1.0f) &
                   (zv >= 0.0f) & (zv <= 1.0f);
        float cx = fminf(fmaxf(xv, 0.0f), 1.0f);
        float cy = fminf(fmaxf(yv, 0.0f), 1.0f);
        float cz = fminf(fmaxf(zv, 0.0f), 1.0f);

        float2 f0, f1, f2, f3;
        trilerp<16>((const float2*)s_emb0, cx, cy, cz, f0);  // LDS gathers
        trilerp<32>(e1, cx, cy, cz, f1);                     // L2 gathers
        trilerp<64>(e2, cx, cy, cz, f2);
        trilerp<128>(e3, cx, cy, cz, f3);

        float m = inb ? 1.0f : 0.0f;
        v4f o0 = { f0.x * m, f0.y * m, f1.x * m, f1.y * m };
        v4f o1 = { f2.x * m, f2.y * m, f3.x * m, f3.y * m };
        v4f* op = (v4f*)(out + (size_t)i * 8);
        __builtin_nontemporal_store(o0, &op[0]);
        __builtin_nontemporal_store(o1, &op[1]);
        __builtin_nontemporal_store(m, &out[(size_t)n * 8 + (size_t)i]);
    }
}

extern "C" void kernel_launch(void* const* d_in, const int* in_sizes, int n_in,
                              void* d_out, int out_size, void* d_ws, size_t ws_size,
                              hipStream_t stream) {
    const float* x    = (const float*)d_in[0];
    const float* emb0 = (const float*)d_in[1];
    const float* emb1 = (const float*)d_in[2];
    const float* emb2 = (const float*)d_in[3];
    const float* emb3 = (const float*)d_in[4];
    float* out = (float*)d_out;
    int n = in_sizes[0] / 3;
    if (n <= 0) return;

    // Persistent-ish grid: caps LDS staging traffic at blocks*32KB while still
    // giving 8 waves/block * 4096 blocks = 32K wave32's of latency hiding.
    int blocks = (n + 255) / 256;
    if (blocks > 4096) blocks = 4096;
    multires_grid_kernel<<<dim3(blocks), dim3(256), 0, stream>>>(
        x, emb0, emb1, emb2, emb3, out, n);
}